// GraphClassifier_6090263626390
// MI455X (gfx1250) — compile-verified
//
#include <hip/hip_runtime.h>

// ---------------------------------------------------------------------------
// 2-layer GCN (PyG GCNConv semantics) for MI455X / gfx1250.
//   deg -> dinv -> norm  |  H = X@W1 (fp32 WMMA)  |  edge scatter (HW f32 atomics)
//   bias+ReLU            |  H2 = H@W2 (fp32 WMMA) |  edge scatter -> out, +b2
// GEMMs use V_WMMA_F32_16X16X4_F32 (full fp32 precision; GEMM is <1% of the
// runtime here -- the job is L2-atomic bound on the 84M scatter adds).
// ---------------------------------------------------------------------------

typedef float v2f __attribute__((ext_vector_type(2)));
typedef float v8f __attribute__((ext_vector_type(8)));

#define N_NODES 50000
#define N_EDGES 1000000
#define IN_DIM  128
#define HID_DIM 64
#define OUT_DIM 16
#define E_TOT   (N_EDGES + N_NODES)   // edges + self loops

// ------------------------------- utilities --------------------------------

__global__ void fill_f32(float* __restrict__ p, float v, int n) {
  int i = blockIdx.x * blockDim.x + threadIdx.x;
  if (i < n) p[i] = v;
}

__global__ void deg_accum(const int* __restrict__ dst, float* __restrict__ deg) {
  int i = blockIdx.x * blockDim.x + threadIdx.x;
  if (i < N_EDGES) unsafeAtomicAdd(&deg[dst[i]], 1.0f);
}

__global__ void deg_to_dinv(float* __restrict__ d) {
  int i = blockIdx.x * blockDim.x + threadIdx.x;
  if (i < N_NODES) {
    float dv = d[i];
    d[i] = dv > 0.0f ? rsqrtf(dv) : 0.0f;
  }
}

__global__ void edge_norm(const int* __restrict__ src, const int* __restrict__ dst,
                          const float* __restrict__ dinv, float* __restrict__ norm) {
  int e = blockIdx.x * blockDim.x + threadIdx.x;
  if (e >= E_TOT) return;
  int s, d;
  if (e < N_EDGES) { s = src[e]; d = dst[e]; } else { s = d = e - N_EDGES; }
  norm[e] = dinv[s] * dinv[d];
}

// ------------------------- fp32 WMMA GEMM (16x16 tile/wave) ----------------
// C[M,N] = A[M,K] @ B[K,N].  blockDim.x == 128 (4 waves), one 16x16 tile/wave.
// A frag (16x4): lane<16 -> row=lane, K pair {k,k+1}; lane>=16 -> K pair {k+2,k+3}
// B frag (4x16): v0 = B[k+2*half][col], v1 = B[k+2*half+1][col]
// C/D: vgpr r -> row = 16*tile + r + 8*half, col = tile*16 + (lane&15)

__global__ void wmma_gemm_f32(const float* __restrict__ A, const float* __restrict__ B,
                              float* __restrict__ C, int M, int K, int N) {
  const int lane = threadIdx.x & 31;
  const int wave = threadIdx.x >> 5;
  const int gw   = blockIdx.x * (blockDim.x >> 5) + wave;
  const int colTiles = N >> 4;
  const int rowTile  = gw / colTiles;
  const int colTile  = gw % colTiles;
  if (rowTile * 16 >= M) return;              // wave-uniform (EXEC stays all-1s)

  const int half = lane >> 4;                 // 0: K{0,1}/M lo rows, 1: K{2,3}/hi
  const int idx  = lane & 15;
  const int arow = rowTile * 16 + idx;
  const int bcol = colTile * 16 + idx;

  v8f acc = {0.f, 0.f, 0.f, 0.f, 0.f, 0.f, 0.f, 0.f};
  const float* __restrict__ arow_p = A + (size_t)arow * K;

  for (int k = 0; k < K; k += 4) {
    v2f a = *(const v2f*)(arow_p + k + 2 * half);        // 8B aligned load
    v2f b;
    b.x = B[(size_t)(k + 2 * half)     * N + bcol];
    b.y = B[(size_t)(k + 2 * half + 1) * N + bcol];
    acc = __builtin_amdgcn_wmma_f32_16x16x4_f32(
        /*neg_a=*/false, a, /*neg_b=*/false, b,
        /*c_mod=*/(short)0, acc, /*reuse_a=*/false, /*reuse_b=*/false);
  }

  const int crow0 = rowTile * 16 + 8 * half;
#pragma unroll
  for (int r = 0; r < 8; ++r)
    C[(size_t)(crow0 + r) * N + bcol] = acc[r];
}

// --------------------- gather * norm -> scatter-add ------------------------
// Layer 1: 64 feats -> 16 threads/edge, float4 each.
__global__ void scatter64(const int* __restrict__ src, const int* __restrict__ dst,
                          const float* __restrict__ norm, const float* __restrict__ H,
                          float* __restrict__ out) {
  int t = blockIdx.x * blockDim.x + threadIdx.x;
  int e = t >> 4, c = t & 15;
  if (e >= E_TOT) return;
  int s, d;
  if (e < N_EDGES) { s = src[e]; d = dst[e]; } else { s = d = e - N_EDGES; }
  float nrm = norm[e];
  const float4 h = *(const float4*)(H + (size_t)s * HID_DIM + c * 4);
  float* o = out + (size_t)d * HID_DIM + c * 4;
  unsafeAtomicAdd(o + 0, h.x * nrm);
  unsafeAtomicAdd(o + 1, h.y * nrm);
  unsafeAtomicAdd(o + 2, h.z * nrm);
  unsafeAtomicAdd(o + 3, h.w * nrm);
}

// Layer 2: 16 feats -> 4 threads/edge, float4 each.
__global__ void scatter16(const int* __restrict__ src, const int* __restrict__ dst,
                          const float* __restrict__ norm, const float* __restrict__ H2,
                          float* __restrict__ out) {
  int t = blockIdx.x * blockDim.x + threadIdx.x;
  int e = t >> 2, c = t & 3;
  if (e >= E_TOT) return;
  int s, d;
  if (e < N_EDGES) { s = src[e]; d = dst[e]; } else { s = d = e - N_EDGES; }
  float nrm = norm[e];
  const float4 h = *(const float4*)(H2 + (size_t)s * OUT_DIM + c * 4);
  float* o = out + (size_t)d * OUT_DIM + c * 4;
  unsafeAtomicAdd(o + 0, h.x * nrm);
  unsafeAtomicAdd(o + 1, h.y * nrm);
  unsafeAtomicAdd(o + 2, h.z * nrm);
  unsafeAtomicAdd(o + 3, h.w * nrm);
}

__global__ void bias_relu64(float* __restrict__ h, const float* __restrict__ b) {
  int i = blockIdx.x * blockDim.x + threadIdx.x;
  if (i < N_NODES * HID_DIM) {
    float v = h[i] + b[i & (HID_DIM - 1)];
    h[i] = v > 0.f ? v : 0.f;
  }
}

__global__ void bias16(float* __restrict__ o, const float* __restrict__ b) {
  int i = blockIdx.x * blockDim.x + threadIdx.x;
  if (i < N_NODES * OUT_DIM) o[i] += b[i & (OUT_DIM - 1)];
}

// ------------------------------- launcher ----------------------------------

extern "C" void kernel_launch(void* const* d_in, const int* in_sizes, int n_in,
                              void* d_out, int out_size, void* d_ws, size_t ws_size,
                              hipStream_t stream) {
  const float* x  = (const float*)d_in[0];
  const int*   ei = (const int*)d_in[1];      // [2, E] int32 (JAX default int)
  const float* W1 = (const float*)d_in[2];
  const float* b1 = (const float*)d_in[3];
  const float* W2 = (const float*)d_in[4];
  const float* b2 = (const float*)d_in[5];
  float* out = (float*)d_out;
  const int* src = ei;
  const int* dst = ei + N_EDGES;

  // workspace carve-out (256B aligned)
  char* ws = (char*)d_ws;
  size_t off = 0;
  auto carve = [&](size_t bytes) -> float* {
    float* p = (float*)(ws + off);
    off += (bytes + 255) & ~(size_t)255;
    return p;
  };
  float* dinv = carve((size_t)N_NODES * 4);            // deg, then deg^-1/2
  float* norm = carve((size_t)E_TOT * 4);              // per-edge norm
  float* H    = carve((size_t)N_NODES * HID_DIM * 4);  // X@W1, reused for H@W2
  float* Hagg = carve((size_t)N_NODES * HID_DIM * 4);  // layer-1 aggregation

  const int B = 256;
  // init (every call: graph-replay safe)
  fill_f32<<<(N_NODES + B - 1) / B, B, 0, stream>>>(dinv, 1.0f, N_NODES); // self loop
  fill_f32<<<(N_NODES * HID_DIM + B - 1) / B, B, 0, stream>>>(Hagg, 0.f, N_NODES * HID_DIM);
  fill_f32<<<(out_size + B - 1) / B, B, 0, stream>>>(out, 0.f, out_size);

  // degrees -> dinv -> per-edge norm
  deg_accum<<<(N_EDGES + B - 1) / B, B, 0, stream>>>(dst, dinv);
  deg_to_dinv<<<(N_NODES + B - 1) / B, B, 0, stream>>>(dinv);
  edge_norm<<<(E_TOT + B - 1) / B, B, 0, stream>>>(src, dst, dinv, norm);

  // layer 1: H = X @ W1 (50000x128x64) -> 3125x4 = 12500 tiles, 4 waves/block
  wmma_gemm_f32<<<12500 / 4, 128, 0, stream>>>(x, W1, H, N_NODES, IN_DIM, HID_DIM);
  scatter64<<<((size_t)E_TOT * 16 + B - 1) / B, B, 0, stream>>>(src, dst, norm, H, Hagg);
  bias_relu64<<<(N_NODES * HID_DIM + B - 1) / B, B, 0, stream>>>(Hagg, b1);

  // layer 2: H2 = Hagg @ W2 (50000x64x16) -> 3125 tiles
  wmma_gemm_f32<<<(3125 + 3) / 4, 128, 0, stream>>>(Hagg, W2, H, N_NODES, HID_DIM, OUT_DIM);
  scatter16<<<((size_t)E_TOT * 4 + B - 1) / B, B, 0, stream>>>(src, dst, norm, H, out);
  bias16<<<(N_NODES * OUT_DIM + B - 1) / B, B, 0, stream>>>(out, b2);
}